// SpMiddleFHD_86517821214985
// MI455X (gfx1250) — compile-verified
//
#include <hip/hip_runtime.h>
#include <hip/hip_bf16.h>
#include <stdint.h>
#include <stddef.h>

// ---------------------------------------------------------------------------
// SpMiddleFHD for MI455X (gfx1250): sparse gather-conv at full resolution
// (30k active voxels), dense masked conv below, channel contraction on
// V_WMMA_F32_16X16X4_F32 (fp32 tensor path, precision-safe).
// Activations channels-last [D][H][W][C]; weights repacked so each lane's
// A and B fragments are single aligned b64 loads with compile-time immediate
// offsets (CIN and COUT are template params). Per tap, all K-fragments are
// staged into register arrays first (batched independent loads -> partial
// loadcnt waits overlap with the WMMA chain). OOB taps read a zero-pad
// region via one pointer select per tap. Workspace requirement: ~215 MB.
// ---------------------------------------------------------------------------

typedef __attribute__((ext_vector_type(2))) float v2f;
typedef __attribute__((ext_vector_type(8))) float v8f;

struct ConvArgs {
  const float* xin; float* xout;
  const float* wf;   // folded weights, layout [tap][cin/4][kh][cout][2]
  const float* bias; // folded BN shift, [cout]
  const float* mask; // per-out-voxel mask (nullptr -> 1)
  const int*   act;  // active out-voxel flat indices (nullptr -> dense)
  const float* zpad; // >=128 floats of zeros (OOB load target)
  int nOut;
  int Di, Hi, Wi, Do, Ho, Wo;
  int sz, sy, sx, pz, py, px, kz, ky, kx;
  int ntilesN;
  int permuteOut;    // final layer: store as [C*D][H][W]
};

// One wave computes a 16-voxel x 16-outchannel tile via implicit GEMM:
//   out[m, n] = sum_{tap} sum_{ci} X[pos(m,tap), ci] * Wf[tap, ci, n]
template <int CIN, int COUT>
__global__ __launch_bounds__(256) void conv3d_wmma_f32(ConvArgs a) {
  constexpr int KCN  = CIN / 4;          // K-chunks of 4 per tap
  constexpr int BKC  = 4 * COUT;         // B floats per kc step
  constexpr int BTAP = KCN * 4 * COUT;   // B floats per tap

  const int lane = threadIdx.x & 31;
  const int wid  = blockIdx.x * (blockDim.x >> 5) + (threadIdx.x >> 5);
  const int tilesM = (a.nOut + 15) >> 4;
  const int tileM  = wid / a.ntilesN;
  const int tileN  = wid - tileM * a.ntilesN;
  if (tileM >= tilesM) return;

  const int m  = lane & 15;   // A-matrix row / C-matrix N index
  const int kh = lane >> 4;   // K-half: this lane holds K = 2*kh, 2*kh+1

  // Decode this lane's output voxel position.
  const int  ovi   = (tileM << 4) + m;
  const bool valid = ovi < a.nOut;
  int flat = 0;
  if (valid) flat = a.act ? a.act[ovi] : ovi;
  const int wo = flat % a.Wo;
  const int t0 = flat / a.Wo;
  const int ho = t0 % a.Ho;
  const int dd = t0 / a.Ho;
  const int iz0 = dd * a.sz - a.pz;
  const int iy0 = ho * a.sy - a.py;
  const int ix0 = wo * a.sx - a.px;

  v8f acc = {0.f, 0.f, 0.f, 0.f, 0.f, 0.f, 0.f, 0.f};
  const int nb = (tileN << 4) + m;          // output channel for B/C lanes
  const float* zp = a.zpad + (kh << 1);
  const float* wtap = a.wf + ((size_t)kh * COUT + nb) * 2;  // tap 0, kc 0

  for (int tz = 0; tz < a.kz; ++tz) {
    const int z = iz0 + tz;
    const bool zok = (unsigned)z < (unsigned)a.Di;
    for (int ty = 0; ty < a.ky; ++ty) {
      const int y = iy0 + ty;
      const bool rowok = valid && zok && ((unsigned)y < (unsigned)a.Hi);
      const int rowIdx = rowok ? (z * a.Hi + y) * a.Wi : 0;
      const float* rowp = a.xin + (size_t)rowIdx * CIN + (kh << 1);
      for (int tx = 0; tx < a.kx; ++tx) {
        const int x = ix0 + tx;
        const bool ok = rowok && ((unsigned)x < (unsigned)a.Wi);
        // Pointer select: OOB lanes stream zeros from zpad (no predication).
        const float* xp = ok ? (rowp + (size_t)x * CIN) : zp;
        // Stage the whole tap: batched independent b64 loads...
        v2f aa[KCN], bb[KCN];
#pragma unroll
        for (int kc = 0; kc < KCN; ++kc) {
          aa[kc] = *(const v2f*)(xp + kc * 4);
          bb[kc] = *(const v2f*)(wtap + kc * BKC);
        }
        // ...then the dependent WMMA chain (partial waits overlap loads).
#pragma unroll
        for (int kc = 0; kc < KCN; ++kc) {
          acc = __builtin_amdgcn_wmma_f32_16x16x4_f32(
              false, aa[kc], false, bb[kc], (short)0, acc, false, false);
        }
        wtap += BTAP;
      }
    }
  }

  // Fused bias + ReLU + mask, scatter per C/D layout (VGPR r -> M=r+8*kh).
  const float bb2 = a.bias[nb];
#pragma unroll
  for (int r = 0; r < 8; ++r) {
    const int m2   = r + (kh << 3);
    const int ovi2 = (tileM << 4) + m2;
    if (ovi2 < a.nOut) {
      const int f2 = a.act ? a.act[ovi2] : ovi2;
      float v = acc[r] + bb2;
      v = v > 0.f ? v : 0.f;
      if (a.mask) v *= a.mask[f2];
      if (a.permuteOut) {
        const int w2 = f2 % a.Wo;
        const int t2 = f2 / a.Wo;
        const int h2 = t2 % a.Ho;
        const int d2 = t2 / a.Ho;
        a.xout[(((size_t)nb * a.Do + d2) * a.Ho + h2) * a.Wo + w2] = v;
      } else {
        a.xout[(size_t)f2 * COUT + nb] = v;
      }
    }
  }
}

// Fold BN into weights: scale = gamma/sqrt(var+eps).
// Repack to [tap][cin/4][kh][cout][2] so a lane's (K=2kh, K=2kh+1) pair for
// output channel n is one contiguous b64.
__global__ void fold_weights(const float* __restrict__ w,
                             const float* __restrict__ gamma,
                             const float* __restrict__ beta,
                             const float* __restrict__ mean,
                             const float* __restrict__ var,
                             float* __restrict__ wf, float* __restrict__ bf,
                             int cout, int cin, int ktot) {
  const int i = blockIdx.x * blockDim.x + threadIdx.x;
  const int total = cout * cin * ktot;
  if (i >= total) return;
  const int co  = i / (cin * ktot);
  const int rem = i - co * (cin * ktot);
  const int ci  = rem / ktot;
  const int tap = rem - ci * ktot;
  const float sc = gamma[co] * (1.0f / sqrtf(var[co] + 1e-3f));
  const int kc = ci >> 2;
  const int kh = (ci >> 1) & 1;
  const int j  = ci & 1;
  const int kcn = cin >> 2;
  wf[(((size_t)(tap * kcn + kc) * 2 + kh) * cout + co) * 2 + j] = w[i] * sc;
  if (ci == 0 && tap == 0) bf[co] = beta[co] - mean[co] * sc;
}

// Scatter voxel features into channels-last full-res grid + dense mask + list
__global__ void scatter_in(const float* __restrict__ feat,
                           const int* __restrict__ coors,
                           float* __restrict__ grid4, float* __restrict__ mask0,
                           int* __restrict__ flat0, int n, int H, int W) {
  const int i = blockIdx.x * blockDim.x + threadIdx.x;
  if (i >= n) return;
  const int z = coors[i * 4 + 1];
  const int y = coors[i * 4 + 2];
  const int x = coors[i * 4 + 3];
  const int f = (z * H + y) * W + x;
  flat0[i] = f;
  mask0[f] = 1.0f;
#pragma unroll
  for (int c = 0; c < 4; ++c) grid4[(size_t)f * 4 + c] = feat[i * 4 + c];
}

// mask_out[v] = any(mask_in in strided 3D window) -- matches ones-conv > 0
__global__ void mask_dilate(const float* __restrict__ mi, float* __restrict__ mo,
                            int Di, int Hi, int Wi, int Do, int Ho, int Wo,
                            int kz, int ky, int kx, int sz, int sy, int sx,
                            int pz, int py, int px) {
  const int i = blockIdx.x * blockDim.x + threadIdx.x;
  if (i >= Do * Ho * Wo) return;
  const int w = i % Wo;
  const int t = i / Wo;
  const int h = t % Ho;
  const int d = t / Ho;
  float any = 0.f;
  for (int tz = 0; tz < kz; ++tz) {
    const int z = d * sz + tz - pz;
    if ((unsigned)z >= (unsigned)Di) continue;
    for (int ty = 0; ty < ky; ++ty) {
      const int y = h * sy + ty - py;
      if ((unsigned)y >= (unsigned)Hi) continue;
      for (int tx = 0; tx < kx; ++tx) {
        const int x = w * sx + tx - px;
        if ((unsigned)x >= (unsigned)Wi) continue;
        if (mi[(z * Hi + y) * Wi + x] != 0.f) any = 1.f;
      }
    }
  }
  mo[i] = any;
}

__global__ void zero_f32(float* __restrict__ p, size_t n) {
  size_t i = (size_t)blockIdx.x * blockDim.x + threadIdx.x;
  const size_t stride = (size_t)gridDim.x * blockDim.x;
  for (; i < n; i += stride) p[i] = 0.f;
}

// ---------------------------------------------------------------------------
struct LC { int sp; int cin, cout; int kz, ky, kx; int sz, sy, sx; int pz, py, px; };

static inline void launch_conv(const ConvArgs& a, int cin, int cout, int blocks,
                               hipStream_t stream) {
  if (cin == 4 && cout == 16)
    conv3d_wmma_f32<4, 16><<<blocks, 256, 0, stream>>>(a);
  else if (cin == 16 && cout == 16)
    conv3d_wmma_f32<16, 16><<<blocks, 256, 0, stream>>>(a);
  else if (cin == 16 && cout == 32)
    conv3d_wmma_f32<16, 32><<<blocks, 256, 0, stream>>>(a);
  else if (cin == 32 && cout == 32)
    conv3d_wmma_f32<32, 32><<<blocks, 256, 0, stream>>>(a);
  else if (cin == 32 && cout == 64)
    conv3d_wmma_f32<32, 64><<<blocks, 256, 0, stream>>>(a);
  else
    conv3d_wmma_f32<64, 64><<<blocks, 256, 0, stream>>>(a);
}

extern "C" void kernel_launch(void* const* d_in, const int* in_sizes, int n_in,
                              void* d_out, int out_size, void* d_ws, size_t ws_size,
                              hipStream_t stream) {
  (void)n_in; (void)out_size; (void)ws_size;
  const float* feat  = (const float*)d_in[0];
  const int*   coors = (const int*)d_in[1];
  const int nvox = in_sizes[0] / 4;

  static const LC L[14] = {
    {0,  4, 16, 3, 3, 3, 1, 1, 1, 1, 1, 1},
    {0, 16, 16, 3, 3, 3, 1, 1, 1, 1, 1, 1},
    {1, 16, 32, 3, 3, 3, 2, 2, 2, 1, 1, 1},
    {0, 32, 32, 3, 3, 3, 1, 1, 1, 1, 1, 1},
    {0, 32, 32, 3, 3, 3, 1, 1, 1, 1, 1, 1},
    {1, 32, 64, 3, 3, 3, 2, 2, 2, 1, 1, 1},
    {0, 64, 64, 3, 3, 3, 1, 1, 1, 1, 1, 1},
    {0, 64, 64, 3, 3, 3, 1, 1, 1, 1, 1, 1},
    {0, 64, 64, 3, 3, 3, 1, 1, 1, 1, 1, 1},
    {1, 64, 64, 3, 3, 3, 2, 2, 2, 0, 1, 1},
    {0, 64, 64, 3, 3, 3, 1, 1, 1, 1, 1, 1},
    {0, 64, 64, 3, 3, 3, 1, 1, 1, 1, 1, 1},
    {0, 64, 64, 3, 3, 3, 1, 1, 1, 1, 1, 1},
    {1, 64, 64, 3, 1, 1, 2, 1, 1, 0, 0, 0},
  };

  const int D0 = 41, H0 = 200, W0 = 176;
  const size_t FULL = (size_t)D0 * H0 * W0;

  float* ws = (float*)d_ws;
  size_t off = 0;
  float* gridA = ws + off; off += FULL * 16;
  float* gridB = ws + off; off += FULL * 16;
  float* grid4 = ws + off; off += FULL * 4;
  float* mask0 = ws + off; off += FULL;
  float* zpad  = ws + off; off += 128;        // stays zero for whole launch
  const size_t zeroCount = off;               // everything above starts 0
  float* masks[5];
  masks[0] = mask0;
  masks[1] = ws + off; off += (size_t)21 * 100 * 88;
  masks[2] = ws + off; off += (size_t)11 * 50 * 44;
  masks[3] = ws + off; off += (size_t)5 * 25 * 22;
  masks[4] = ws + off; off += (size_t)2 * 25 * 22;
  int* flat0 = (int*)(ws + off); off += (size_t)nvox;

  float* wfold[14];
  float* bfold[14];
  for (int i = 0; i < 14; ++i) {
    const int kt = L[i].kz * L[i].ky * L[i].kx;
    wfold[i] = ws + off; off += (size_t)kt * L[i].cin * L[i].cout;
  }
  for (int i = 0; i < 14; ++i) { bfold[i] = ws + off; off += (size_t)L[i].cout; }

  // 1) zero the full-res grids + dense mask + zpad
  zero_f32<<<4096, 256, 0, stream>>>(ws, zeroCount);

  // 2) fold BN into weights for all layers
  for (int i = 0; i < 14; ++i) {
    const int kt = L[i].kz * L[i].ky * L[i].kx;
    const int total = L[i].cout * L[i].cin * kt;
    const float* w     = (const float*)d_in[3 + 5 * i + 0];
    const float* gamma = (const float*)d_in[3 + 5 * i + 1];
    const float* beta  = (const float*)d_in[3 + 5 * i + 2];
    const float* mean  = (const float*)d_in[3 + 5 * i + 3];
    const float* var   = (const float*)d_in[3 + 5 * i + 4];
    fold_weights<<<(total + 255) / 256, 256, 0, stream>>>(
        w, gamma, beta, mean, var, wfold[i], bfold[i], L[i].cout, L[i].cin, kt);
  }

  // 3) scatter input voxels
  scatter_in<<<(nvox + 255) / 256, 256, 0, stream>>>(feat, coors, grid4, mask0,
                                                     flat0, nvox, H0, W0);

  // 4) layer pipeline
  int curD = D0, curH = H0, curW = W0;
  int mlevel = 0;
  for (int i = 0; i < 14; ++i) {
    const LC& c = L[i];
    ConvArgs a;
    a.Di = curD; a.Hi = curH; a.Wi = curW;
    int Do = curD, Ho = curH, Wo = curW;
    if (c.sp) {
      Do = (curD + 2 * c.pz - c.kz) / c.sz + 1;
      Ho = (curH + 2 * c.py - c.ky) / c.sy + 1;
      Wo = (curW + 2 * c.px - c.kx) / c.sx + 1;
      const int total = Do * Ho * Wo;
      mask_dilate<<<(total + 255) / 256, 256, 0, stream>>>(
          masks[mlevel], masks[mlevel + 1], curD, curH, curW, Do, Ho, Wo,
          c.kz, c.ky, c.kx, c.sz, c.sy, c.sx, c.pz, c.py, c.px);
      ++mlevel;
    }
    a.Do = Do; a.Ho = Ho; a.Wo = Wo;
    a.sz = c.sz; a.sy = c.sy; a.sx = c.sx;
    a.pz = c.pz; a.py = c.py; a.px = c.px;
    a.kz = c.kz; a.ky = c.ky; a.kx = c.kx;
    a.xin  = (i == 0) ? grid4 : ((i & 1) ? gridA : gridB);
    a.xout = (i == 13) ? (float*)d_out : ((i & 1) ? gridB : gridA);
    a.wf = wfold[i]; a.bias = bfold[i];
    a.zpad = zpad;
    a.permuteOut = (i == 13) ? 1 : 0;
    if (mlevel == 0) {           // full-res subm layers: sparse active list
      a.act = flat0; a.mask = nullptr; a.nOut = nvox;
    } else {                     // lower res: dense grid with mask multiply
      a.act = nullptr; a.mask = masks[mlevel]; a.nOut = Do * Ho * Wo;
    }
    a.ntilesN = c.cout / 16;
    const int tilesM = (a.nOut + 15) / 16;
    const int waves  = tilesM * a.ntilesN;
    const int blocks = (waves + 7) / 8;    // 256 threads = 8 waves/block
    launch_conv(a, c.cin, c.cout, blocks, stream);
    curD = Do; curH = Ho; curW = Wo;
  }
}